// PScanTriton_47708496724083
// MI455X (gfx1250) — compile-verified
//
#include <hip/hip_runtime.h>
#include <stdint.h>

// ---------------------------------------------------------------------------
// Complex linear-recurrence scan  y_t = a_t*y_{t-1} + x_t  over L, per (b,d).
// B=8, L=2048, D=512, complex64 (separate re/im planes), out = [B,L,D,2] f32.
//
// Memory-bound (201 MB min traffic ~ 8.6us @ 23.3 TB/s). 3-phase chunked scan
// for L-parallelism; inputs staged to LDS with the CDNA5 Tensor Data Mover
// (tensor_load_to_lds + s_wait_tensorcnt), double-buffered.
// ---------------------------------------------------------------------------

static constexpr int B_ = 8;
static constexpr int L_ = 2048;
static constexpr int D_ = 512;

static constexpr int NCHUNK = 32;          // chunks along L
static constexpr int T_     = L_ / NCHUNK; // 64 timesteps per chunk
static constexpr int BW     = 256;         // d-lanes per block (= blockDim.x)
static constexpr int ROWS   = 8;           // timesteps per TDM stage
static constexpr int NSTAGE = T_ / ROWS;   // 8 stages per chunk

// ---------------------------------------------------------------------------
// TDM availability (compile-time). Falls back to coalesced global loads.
// ---------------------------------------------------------------------------
#if defined(__HIP_DEVICE_COMPILE__) && \
    __has_builtin(__builtin_amdgcn_tensor_load_to_lds) && \
    __has_builtin(__builtin_amdgcn_s_wait_tensorcnt)
#define USE_TDM 1
#else
#define USE_TDM 0
#endif

#if USE_TDM
typedef unsigned int v4u __attribute__((ext_vector_type(4)));
typedef int          v8i __attribute__((ext_vector_type(8)));
typedef int          v4i __attribute__((ext_vector_type(4)));

// Issue one TDM 2-D tile load: tileW x tileH f32 elements, row stride
// `strideElems`, from global `gsrc` into LDS byte offset `lds_byte`.
// D# packing per CDNA5 ISA.
__device__ __forceinline__ void tdm_tile_load(const float* gsrc,
                                              unsigned lds_byte,
                                              unsigned tileW, unsigned tileH,
                                              unsigned strideElems) {
  unsigned long long ga = (unsigned long long)(uintptr_t)gsrc;
  v4u g0;
  g0[0] = 1u;                                            // count=1 (valid), user mode
  g0[1] = lds_byte;                                      // LDS dest byte offset
  g0[2] = (unsigned)(ga & 0xFFFFFFFFull);                // global_addr[31:0]
  g0[3] = (unsigned)((ga >> 32) & 0x01FFFFFFull)         // global_addr[56:32]
        | (2u << 30);                                    // type = 2 ("image")
  const unsigned td0 = strideElems;                      // tensor_dim0 >= tileW
  const unsigned td1 = 0xFFFFu;                          // tensor_dim1 >= tileH
  v8i g1;
  g1[0] = 0x00020000;                                    // data_size=2 (4B); mask=0
  g1[1] = (int)((td0 & 0xFFFFu) << 16);                  // barrier_addr=0 | dim0.lo16
  g1[2] = (int)(((td0 >> 16) & 0xFFFFu) |
                ((td1 & 0xFFFFu) << 16));                // dim0.hi16 | dim1.lo16
  g1[3] = (int)(((td1 >> 16) & 0xFFFFu) |
                ((tileW & 0xFFFFu) << 16));              // dim1.hi16 | tile_dim0
  g1[4] = (int)(tileH & 0xFFFFu);                        // tile_dim1 | tile_dim2=0
  g1[5] = (int)strideElems;                              // tensor_dim0_stride[31:0]
  g1[6] = 0;                                             // stride0.hi16 | stride1.lo16
  g1[7] = 0;
  v4i gz = {0, 0, 0, 0};                                 // groups 2/3 unused (2-D)
#if __clang_major__ >= 23
  v8i gz8 = {0, 0, 0, 0, 0, 0, 0, 0};
  __builtin_amdgcn_tensor_load_to_lds(g0, g1, gz, gz, gz8, 0);
#else
  __builtin_amdgcn_tensor_load_to_lds(g0, g1, gz, gz, 0);
#endif
}

__device__ __forceinline__ void issue_stage(const float* Ar, const float* Ai,
                                            const float* Xr, const float* Xi,
                                            size_t gofs,
                                            float (*buf)[ROWS * BW]) {
  tdm_tile_load(Ar + gofs, (unsigned)(uintptr_t)&buf[0][0], BW, ROWS, D_);
  tdm_tile_load(Ai + gofs, (unsigned)(uintptr_t)&buf[1][0], BW, ROWS, D_);
  tdm_tile_load(Xr + gofs, (unsigned)(uintptr_t)&buf[2][0], BW, ROWS, D_);
  tdm_tile_load(Xi + gofs, (unsigned)(uintptr_t)&buf[3][0], BW, ROWS, D_);
}
#endif  // USE_TDM

// combine (complex): running (aP,xP) <- ( a*aP , a*xP + x )
#define CPX_STEP(ar, ai, xr, xi, pAr, pAi, pXr, pXi)        \
  {                                                         \
    float nAr = (ar) * (pAr) - (ai) * (pAi);                \
    float nAi = (ar) * (pAi) + (ai) * (pAr);                \
    float nXr = (ar) * (pXr) - (ai) * (pXi) + (xr);         \
    float nXi = (ar) * (pXi) + (ai) * (pXr) + (xi);         \
    pAr = nAr; pAi = nAi; pXr = nXr; pXi = nXi;             \
  }

// ---------------------------------------------------------------------------
// Phase 1: per-(b,d,chunk) aggregate (prod a, local scan end).
// ---------------------------------------------------------------------------
__global__ __launch_bounds__(BW) void pscan_phase1(const float* __restrict__ Ar,
                                                   const float* __restrict__ Ai,
                                                   const float* __restrict__ Xr,
                                                   const float* __restrict__ Xi,
                                                   float4* __restrict__ agg) {
  const int tid = threadIdx.x;
  const int d0  = blockIdx.x * BW;
  const int c   = blockIdx.y;
  const int b   = blockIdx.z;
  const size_t base = ((size_t)b * L_ + (size_t)c * T_) * D_ + d0;

  float pAr = 1.f, pAi = 0.f, pXr = 0.f, pXi = 0.f;

#if USE_TDM
  __shared__ float sm[2][4][ROWS * BW];  // 64 KB, double-buffered A/X tiles
  const unsigned wid =
      ((unsigned)__builtin_amdgcn_readfirstlane((int)threadIdx.x)) >> 5;
  if (wid == 0) issue_stage(Ar, Ai, Xr, Xi, base, sm[0]);
  for (int s = 0; s < NSTAGE; ++s) {
    const int buf = s & 1;
    if (wid == 0) {
      if (s + 1 < NSTAGE) {
        issue_stage(Ar, Ai, Xr, Xi, base + (size_t)(s + 1) * ROWS * D_,
                    sm[(s + 1) & 1]);
        __builtin_amdgcn_s_wait_tensorcnt(4);  // oldest 4 (this stage) done
      } else {
        __builtin_amdgcn_s_wait_tensorcnt(0);
      }
    }
    __syncthreads();
#pragma unroll
    for (int r = 0; r < ROWS; ++r) {
      float ar = sm[buf][0][r * BW + tid];
      float ai = sm[buf][1][r * BW + tid];
      float xr = sm[buf][2][r * BW + tid];
      float xi = sm[buf][3][r * BW + tid];
      CPX_STEP(ar, ai, xr, xi, pAr, pAi, pXr, pXi);
    }
    __syncthreads();
  }
#else
  for (int t = 0; t < T_; ++t) {
    size_t i = base + (size_t)t * D_ + tid;
    CPX_STEP(Ar[i], Ai[i], Xr[i], Xi[i], pAr, pAi, pXr, pXi);
  }
#endif

  agg[(b * NCHUNK + c) * D_ + d0 + tid] = make_float4(pAr, pAi, pXr, pXi);
}

// ---------------------------------------------------------------------------
// Phase 2: scan the NCHUNK aggregates per channel -> carry into each chunk.
// ---------------------------------------------------------------------------
__global__ __launch_bounds__(256) void pscan_phase2(const float4* __restrict__ agg,
                                                    float2* __restrict__ carry) {
  const int g = blockIdx.x * blockDim.x + threadIdx.x;  // 0..B*D-1
  const int b = g / D_;
  const int d = g % D_;
  float rAr = 1.f, rAi = 0.f, rXr = 0.f, rXi = 0.f;
  for (int c = 0; c < NCHUNK; ++c) {
    const int idx = (b * NCHUNK + c) * D_ + d;
    carry[idx] = make_float2(rXr, rXi);  // y entering chunk c
    float4 a = agg[idx];
    CPX_STEP(a.x, a.y, a.z, a.w, rAr, rAi, rXr, rXi);
  }
}

// ---------------------------------------------------------------------------
// Phase 3: re-run each chunk's recurrence starting from its carry; write out.
// ---------------------------------------------------------------------------
__global__ __launch_bounds__(BW) void pscan_phase3(const float* __restrict__ Ar,
                                                   const float* __restrict__ Ai,
                                                   const float* __restrict__ Xr,
                                                   const float* __restrict__ Xi,
                                                   const float2* __restrict__ carry,
                                                   float2* __restrict__ out) {
  const int tid = threadIdx.x;
  const int d0  = blockIdx.x * BW;
  const int c   = blockIdx.y;
  const int b   = blockIdx.z;
  const size_t base = ((size_t)b * L_ + (size_t)c * T_) * D_ + d0;

  float2 cv = carry[(b * NCHUNK + c) * D_ + d0 + tid];
  float yr = cv.x, yi = cv.y;

#if USE_TDM
  __shared__ float sm[2][4][ROWS * BW];
  const unsigned wid =
      ((unsigned)__builtin_amdgcn_readfirstlane((int)threadIdx.x)) >> 5;
  if (wid == 0) issue_stage(Ar, Ai, Xr, Xi, base, sm[0]);
  for (int s = 0; s < NSTAGE; ++s) {
    const int buf = s & 1;
    if (wid == 0) {
      if (s + 1 < NSTAGE) {
        issue_stage(Ar, Ai, Xr, Xi, base + (size_t)(s + 1) * ROWS * D_,
                    sm[(s + 1) & 1]);
        __builtin_amdgcn_s_wait_tensorcnt(4);
      } else {
        __builtin_amdgcn_s_wait_tensorcnt(0);
      }
    }
    __syncthreads();
#pragma unroll
    for (int r = 0; r < ROWS; ++r) {
      float ar = sm[buf][0][r * BW + tid];
      float ai = sm[buf][1][r * BW + tid];
      float xr = sm[buf][2][r * BW + tid];
      float xi = sm[buf][3][r * BW + tid];
      float nyr = ar * yr - ai * yi + xr;
      float nyi = ar * yi + ai * yr + xi;
      yr = nyr; yi = nyi;
      out[base + (size_t)(s * ROWS + r) * D_ + tid] = make_float2(yr, yi);
    }
    __syncthreads();
  }
#else
  for (int t = 0; t < T_; ++t) {
    size_t i = base + (size_t)t * D_ + tid;
    float ar = Ar[i], ai = Ai[i], xr = Xr[i], xi = Xi[i];
    float nyr = ar * yr - ai * yi + xr;
    float nyi = ar * yi + ai * yr + xi;
    yr = nyr; yi = nyi;
    out[i] = make_float2(yr, yi);
  }
#endif
}

// ---------------------------------------------------------------------------
// Fallback single-pass kernel (used only if workspace is too small).
// ---------------------------------------------------------------------------
__global__ __launch_bounds__(256) void pscan_naive(const float* __restrict__ Ar,
                                                   const float* __restrict__ Ai,
                                                   const float* __restrict__ Xr,
                                                   const float* __restrict__ Xi,
                                                   float2* __restrict__ out) {
  const int g = blockIdx.x * blockDim.x + threadIdx.x;  // 0..B*D-1
  const int b = g / D_;
  const int d = g % D_;
  float yr = 0.f, yi = 0.f;
  for (int t = 0; t < L_; ++t) {
    size_t i = ((size_t)b * L_ + t) * D_ + d;
    float ar = Ar[i], ai = Ai[i], xr = Xr[i], xi = Xi[i];
    float nyr = ar * yr - ai * yi + xr;
    float nyi = ar * yi + ai * yr + xi;
    yr = nyr; yi = nyi;
    out[i] = make_float2(yr, yi);
  }
}

// ---------------------------------------------------------------------------
extern "C" void kernel_launch(void* const* d_in, const int* in_sizes, int n_in,
                              void* d_out, int out_size, void* d_ws,
                              size_t ws_size, hipStream_t stream) {
  (void)in_sizes; (void)n_in; (void)out_size;
  const float* Ar = (const float*)d_in[0];
  const float* Ai = (const float*)d_in[1];
  const float* Xr = (const float*)d_in[2];
  const float* Xi = (const float*)d_in[3];
  float2* out = (float2*)d_out;

  const size_t nAgg     = (size_t)B_ * NCHUNK * D_;
  const size_t aggBytes = nAgg * sizeof(float4);   // 2 MB
  const size_t carBytes = nAgg * sizeof(float2);   // 1 MB

  if (ws_size >= aggBytes + carBytes) {
    float4* agg = (float4*)d_ws;
    float2* car = (float2*)((char*)d_ws + aggBytes);
    dim3 blk(BW);
    dim3 grd(D_ / BW, NCHUNK, B_);  // 2 x 32 x 8 = 512 blocks
    pscan_phase1<<<grd, blk, 0, stream>>>(Ar, Ai, Xr, Xi, agg);
    pscan_phase2<<<(B_ * D_) / 256, 256, 0, stream>>>(agg, car);
    pscan_phase3<<<grd, blk, 0, stream>>>(Ar, Ai, Xr, Xi, car, out);
  } else {
    pscan_naive<<<(B_ * D_) / 256, 256, 0, stream>>>(Ar, Ai, Xr, Xi, out);
  }
}